// Model_46789373723090
// MI455X (gfx1250) — compile-verified
//
#include <hip/hip_runtime.h>

#define RES    128
#define N_PTS  (RES * RES)   // 16384 queries == 16384 candidates
#define NCH    256
#define NB     2
#define NPB    6             // batch * plane combos

#define TILES_PER_WAVE   2
#define WAVES_PER_BLOCK  8
#define BLOCK_THREADS    (WAVES_PER_BLOCK * 32)           // 256
#define TILES_PER_BLOCK  (WAVES_PER_BLOCK * TILES_PER_WAVE) // 16
#define BLOCKS_PER_PB    ((N_PTS / 16) / TILES_PER_BLOCK)   // 64
#define STAGE_CAND       2048                              // candidates per LDS stage (32 KB)
#define N_STAGES         (N_PTS / STAGE_CAND)              // 8
#define CHUNKS_PER_STAGE (STAGE_CAND / 16)                 // 128

typedef __attribute__((ext_vector_type(2))) float v2f;
typedef __attribute__((ext_vector_type(8))) float v8f;

#if defined(__has_builtin)
#if __has_builtin(__builtin_amdgcn_global_load_async_to_lds_b128)
#define HAVE_ASYNC_LDS 1
#endif
#endif

#ifdef HAVE_ASYNC_LDS
// Parameter types per hipcc diagnostic: AS1 pointer ("__device__") to
// int __attribute__((vector_size(16))), and LDS side in AS3.
typedef int v4i_raw __attribute__((__vector_size__(4 * sizeof(int))));
typedef __attribute__((address_space(1))) v4i_raw* gas_v4i;
typedef __attribute__((address_space(3))) v4i_raw* las_v4i;
#endif

// ---------------------------------------------------------------------------
// Kernel 1: build candidate arrays per (batch, plane):
//   q[pb][m] = (p0, p1, -0.5*(p0^2+p1^2), 0)   where (p0,p1) = 2 * projected pcd
// ---------------------------------------------------------------------------
__global__ void prep_kernel(const float* __restrict__ pcd, float4* __restrict__ q) {
  int t = blockIdx.x * blockDim.x + threadIdx.x;
  if (t >= NB * N_PTS) return;
  int b = t / N_PTS, m = t - b * N_PTS;
  const float* p = pcd + (size_t)(b * N_PTS + m) * 3;
  float x = 2.0f * p[0], y = 2.0f * p[1], z = 2.0f * p[2];
  // plane 0: (x, y); plane 1: (x, z); plane 2: (z, y)
  q[(size_t)(b * 3 + 0) * N_PTS + m] = make_float4(x, y, -0.5f * (x * x + y * y), 0.0f);
  q[(size_t)(b * 3 + 1) * N_PTS + m] = make_float4(x, z, -0.5f * (x * x + z * z), 0.0f);
  q[(size_t)(b * 3 + 2) * N_PTS + m] = make_float4(z, y, -0.5f * (z * z + y * y), 0.0f);
}

// ---------------------------------------------------------------------------
// Kernel 2: WMMA nearest-neighbor argmin with async LDS staging.
// Block = 8 waves; each wave owns 2 query tiles (32 queries).  Per stage the
// block async-copies 2048 candidates (32 KB) to LDS, then every wave runs
// 128 chunks: one ds_load_b128 feeds two v_wmma_f32_16x16x4_f32.
// score[n,m] = u.p - 0.5||p||^2  (argmax == argmin of distance).
// ---------------------------------------------------------------------------
__global__ __launch_bounds__(BLOCK_THREADS) void argmin_kernel(
    const float4* __restrict__ q, int* __restrict__ idx_out) {
  __shared__ float4 smem[STAGE_CAND];   // 32 KB

  const int  tid  = threadIdx.x;
  const int  lane = tid & 31;
  const int  wave = tid >> 5;
  const int  col  = lane & 15;
  const bool hi   = lane >= 16;

  const int pb    = blockIdx.x / BLOCKS_PER_PB;
  const int blk   = blockIdx.x % BLOCKS_PER_PB;
  const int tile0 = blk * TILES_PER_BLOCK + wave * TILES_PER_WAVE; // wave's first tile

  // A matrices (loop-invariant): row n -> (u0, u1, 1, 0)
  // Layout 16x4 f32: lanes 0-15 hold K0/K1 in VGPR0/1, lanes 16-31 hold K2/K3.
  const float step = 2.0f / (float)(RES - 1);
  v2f a0, a1;
  {
    int qn0 = (tile0 + 0) * 16 + col;
    int qn1 = (tile0 + 1) * 16 + col;
    float u0x = -1.0f + step * (float)(qn0 & (RES - 1));
    float u0y = -1.0f + step * (float)(qn0 >> 7);
    float u1x = -1.0f + step * (float)(qn1 & (RES - 1));
    float u1y = -1.0f + step * (float)(qn1 >> 7);
    a0.x = hi ? 1.0f : u0x;  a0.y = hi ? 0.0f : u0y;
    a1.x = hi ? 1.0f : u1x;  a1.y = hi ? 0.0f : u1y;
  }

  const float4* __restrict__ qp = q + (size_t)pb * N_PTS;

  float best0[8], best1[8];
  int   bidx0[8], bidx1[8];
#pragma unroll
  for (int r = 0; r < 8; ++r) {
    best0[r] = -3.4e38f; best1[r] = -3.4e38f;
    bidx0[r] = 0;        bidx1[r] = 0;
  }

  for (int stage = 0; stage < N_STAGES; ++stage) {
    // ---- cooperative fill: 2048 float4 by 256 threads (8 each, no divergence)
    const float4* src = qp + stage * STAGE_CAND;
#ifdef HAVE_ASYNC_LDS
#pragma unroll
    for (int i = tid; i < STAGE_CAND; i += BLOCK_THREADS) {
      __builtin_amdgcn_global_load_async_to_lds_b128(
          (gas_v4i)(float4*)(src + i), (las_v4i)(&smem[i]), 0, 0);
    }
#if __has_builtin(__builtin_amdgcn_s_wait_asynccnt)
    __builtin_amdgcn_s_wait_asynccnt(0);
#else
    asm volatile("s_wait_asynccnt 0x0" ::: "memory");
#endif
#else
#pragma unroll
    for (int i = tid; i < STAGE_CAND; i += BLOCK_THREADS) smem[i] = src[i];
#endif
    __syncthreads();

    // ---- consume: one LDS b128 load feeds two WMMAs
#pragma unroll 2
    for (int cc = 0; cc < CHUNKS_PER_STAGE; ++cc) {
      const int chunk = stage * CHUNKS_PER_STAGE + cc;
      float4 v = smem[cc * 16 + col];
      v2f b;
      b.x = hi ? v.z : v.x;   // lanes 0-15: (p0,p1); lanes 16-31: (-0.5||p||^2, 0)
      b.y = hi ? 0.0f : v.y;

      v8f c = {};
      v8f d0 = __builtin_amdgcn_wmma_f32_16x16x4_f32(
          false, a0, false, b, (short)0, c, false, false);
      v8f d1 = __builtin_amdgcn_wmma_f32_16x16x4_f32(
          false, a1, false, b, (short)0, c, false, false);

#pragma unroll
      for (int r = 0; r < 8; ++r) {
        float s0 = d0[r];
        if (s0 > best0[r]) { best0[r] = s0; bidx0[r] = chunk; } // strict > keeps earliest
        float s1 = d1[r];
        if (s1 > best1[r]) { best1[r] = s1; bidx1[r] = chunk; }
      }
    }
    __syncthreads();   // LDS reuse across stages
  }

  // ---- reduce across the 16 lanes holding each query row; tie-break min index
#pragma unroll
  for (int t = 0; t < TILES_PER_WAVE; ++t) {
    const float* bestp = t ? best1 : best0;
    const int*   bidxp = t ? bidx1 : bidx0;
#pragma unroll
    for (int r = 0; r < 8; ++r) {
      float s  = bestp[r];
      int   gi = bidxp[r] * 16 + col;   // full candidate index
#pragma unroll
      for (int off = 1; off < 16; off <<= 1) {
        float os  = __shfl_xor(s, off, 32);
        int   ogi = __shfl_xor(gi, off, 32);
        if (os > s || (os == s && ogi < gi)) { s = os; gi = ogi; }
      }
      if (col == 0) {
        int row = r + (hi ? 8 : 0);
        idx_out[(size_t)pb * N_PTS + (tile0 + t) * 16 + row] = gi;
      }
    }
  }
}

// ---------------------------------------------------------------------------
// Kernel 3: feature gather.  out[b,p,c,n] = img_feat[b,c,idx[b,p,n]]
// Coalesced 4B writes; gathered reads land in L2 (img_feat fully resident).
// ---------------------------------------------------------------------------
__global__ void gather_kernel(const float* __restrict__ feat,
                              const int* __restrict__ idx,
                              float* __restrict__ out) {
  int n   = blockIdx.x * blockDim.x + threadIdx.x;  // 0..16383
  int row = blockIdx.y;                             // (b*3+p)*256 + c, 0..1535
  int bp  = row >> 8;
  int c   = row & 255;
  int b   = bp / 3;
  int j   = idx[(size_t)bp * N_PTS + n];
  out[(size_t)row * N_PTS + n] = feat[(size_t)(b * NCH + c) * N_PTS + j];
}

// ---------------------------------------------------------------------------
extern "C" void kernel_launch(void* const* d_in, const int* in_sizes, int n_in,
                              void* d_out, int out_size, void* d_ws, size_t ws_size,
                              hipStream_t stream) {
  const float* pcd  = (const float*)d_in[0];   // [2, 16384, 3]
  const float* feat = (const float*)d_in[1];   // [2, 256, 128, 128]
  float*       out  = (float*)d_out;           // [2, 3, 256, 128, 128]

  // Workspace layout: candidate float4 arrays, then argmin indices.
  float4* q   = (float4*)d_ws;                                     // 6*16384*16 B = 1.5 MB
  int*    idx = (int*)((char*)d_ws + (size_t)NPB * N_PTS * 16);    // 6*16384*4  B

  int nprep = NB * N_PTS;
  prep_kernel<<<(nprep + 255) / 256, 256, 0, stream>>>(pcd, q);

  argmin_kernel<<<NPB * BLOCKS_PER_PB, BLOCK_THREADS, 0, stream>>>(q, idx);

  dim3 ggrid(N_PTS / 256, NPB * NCH);
  gather_kernel<<<ggrid, 256, 0, stream>>>(feat, idx, out);
}